// Decoder_23648089932279
// MI455X (gfx1250) — compile-verified
//
#include <hip/hip_runtime.h>
#include <math.h>

// ---------------------------------------------------------------------------
// Problem constants
// ---------------------------------------------------------------------------
constexpr int B    = 16;
constexpr int S    = 128;
constexpr int V    = 32000;
constexpr int EEMB = 512;
constexpr int ENC2 = 1024;
constexpr int H    = 1024;
constexpr int A    = 1024;
constexpr int XDIM = EEMB + ENC2;   // 1536, GRU input dim
constexpr int TSTEPS = S - 1;       // 127 recurrent steps

typedef float v8f __attribute__((ext_vector_type(8)));
typedef float f2  __attribute__((ext_vector_type(2)));

// One V_WMMA_F32_16X16X4_F32 step: D = A(16x4) * B(4x16) + C, f32 throughout.
__device__ __forceinline__ v8f wmma4(f2 a, f2 b, v8f c) {
    return __builtin_amdgcn_wmma_f32_16x16x4_f32(
        /*neg_a=*/false, a, /*neg_b=*/false, b,
        /*c_mod=*/(short)0, c, /*reuse_a=*/false, /*reuse_b=*/false);
}

// ---------------------------------------------------------------------------
// Kernel 1: enc_proj[b,s,a] = encoder_outs[b,s,:] @ W_enc[:,a]
// M = B*S = 2048, K = ENC2 = 1024, N = A = 1024.
// Register-blocked: one wave computes 4 M-tiles (64x16 of C), so each
// strided W_enc load feeds 4 independent WMMAs (4x less B traffic).
// ---------------------------------------------------------------------------
__global__ void __launch_bounds__(256)
encproj_kernel(const float* __restrict__ enc, const float* __restrict__ Wenc,
               float* __restrict__ encp) {
    const int lane  = threadIdx.x & 31;
    const int warp  = threadIdx.x >> 5;
    const int grp   = blockIdx.x * 8 + warp;          // 32 * 64 = 2048 groups
    const int mg    = grp >> 6;                       // M group (4 tiles each)
    const int tN    = grp & 63;
    const int nl    = lane & 15;
    const int koff  = (lane < 16) ? 0 : 2;
    const int n     = tN * 16 + nl;

    const float* A0 = enc + (size_t)(mg * 64 +  0 + nl) * ENC2;
    const float* A1 = enc + (size_t)(mg * 64 + 16 + nl) * ENC2;
    const float* A2 = enc + (size_t)(mg * 64 + 32 + nl) * ENC2;
    const float* A3 = enc + (size_t)(mg * 64 + 48 + nl) * ENC2;

    v8f c0 = {}, c1 = {}, c2 = {}, c3 = {};
    for (int k = 0; k < ENC2; k += 4) {
        f2 b = { Wenc[(size_t)(k + koff)     * A + n],
                 Wenc[(size_t)(k + koff + 1) * A + n] };
        c0 = wmma4(*(const f2*)(A0 + k + koff), b, c0);
        c1 = wmma4(*(const f2*)(A1 + k + koff), b, c1);
        c2 = wmma4(*(const f2*)(A2 + k + koff), b, c2);
        c3 = wmma4(*(const f2*)(A3 + k + koff), b, c3);
    }
    const int mh = (lane < 16) ? 0 : 8;
    for (int i = 0; i < 8; ++i) {
        encp[(size_t)(mg * 64 +  0 + i + mh) * A + n] = c0[i];
        encp[(size_t)(mg * 64 + 16 + i + mh) * A + n] = c1[i];
        encp[(size_t)(mg * 64 + 32 + i + mh) * A + n] = c2[i];
        encp[(size_t)(mg * 64 + 48 + i + mh) * A + n] = c3[i];
    }
}

// ---------------------------------------------------------------------------
// Kernel 2: out[:,0,:] = one-hot(0)   (d_out is poisoned, must be written)
// ---------------------------------------------------------------------------
__global__ void row0_kernel(float* __restrict__ out) {
    int idx = blockIdx.x * blockDim.x + threadIdx.x;
    if (idx >= B * V) return;
    int b = idx / V, v = idx % V;
    out[(size_t)b * S * V + v] = (v == 0) ? 1.0f : 0.0f;
}

// ---------------------------------------------------------------------------
// Kernel 3 (per step): attention.  One workgroup per batch row.
//   hW = h @ W_dec; sc[s] = v_attn . tanh(enc_proj[b,s,:] + hW); softmax; ctx.
// All operands L2-resident (W_dec 4MB, enc_proj 8.4MB, encoder_outs 8.4MB).
// ---------------------------------------------------------------------------
__global__ void __launch_bounds__(256)
attn_kernel(const float* __restrict__ hprev, const float* __restrict__ Wdec,
            const float* __restrict__ vattn, const float* __restrict__ encp,
            const float* __restrict__ enc,  const int* __restrict__ lengths,
            float* __restrict__ ctx) {
    __shared__ float sh_h[H];
    __shared__ float sh_hw[A];
    __shared__ float sh_sc[S];
    __shared__ float sh_e[S];
    __shared__ float sh_red[2];

    const int b   = blockIdx.x;
    const int tid = threadIdx.x;

    for (int i = tid; i < H; i += 256) sh_h[i] = hprev[(size_t)b * H + i];
    __syncthreads();

    // hW[a] = sum_h h[h] * W_dec[h, a]   (4 outputs per thread, coalesced B)
    float acc0 = 0.f, acc1 = 0.f, acc2 = 0.f, acc3 = 0.f;
    for (int h = 0; h < H; ++h) {
        float hv = sh_h[h];
        const float* wr = Wdec + (size_t)h * A + tid;
        acc0 = fmaf(hv, wr[0],   acc0);
        acc1 = fmaf(hv, wr[256], acc1);
        acc2 = fmaf(hv, wr[512], acc2);
        acc3 = fmaf(hv, wr[768], acc3);
    }
    sh_hw[tid]       = acc0;
    sh_hw[tid + 256] = acc1;
    sh_hw[tid + 512] = acc2;
    sh_hw[tid + 768] = acc3;
    __syncthreads();

    // scores
    const int len = lengths[b];
    if (tid < S) {
        float sc = -INFINITY;
        if (tid < len) {
            const float* ep = encp + ((size_t)b * S + tid) * A;
            float a0 = 0.f;
            for (int a = 0; a < A; ++a)
                a0 = fmaf(tanhf(ep[a] + sh_hw[a]), vattn[a], a0);
            sc = a0;
        }
        sh_sc[tid] = sc;
    }
    __syncthreads();

    if (tid == 0) {
        float mx = sh_sc[0];
        for (int s = 1; s < S; ++s) mx = fmaxf(mx, sh_sc[s]);
        sh_red[0] = mx;
    }
    __syncthreads();
    if (tid < S) sh_e[tid] = expf(sh_sc[tid] - sh_red[0]);   // -inf -> 0
    __syncthreads();
    if (tid == 0) {
        float sum = 0.f;
        for (int s = 0; s < S; ++s) sum += sh_e[s];
        sh_red[1] = sum;
    }
    __syncthreads();

    // ctx[b,d] = (1/sum) * sum_s e[s] * enc[b,s,d]
    const float inv = 1.0f / sh_red[1];
    for (int d = tid; d < ENC2; d += 256) {
        float a0 = 0.f;
        const float* er = enc + (size_t)b * S * ENC2 + d;
        for (int s = 0; s < S; ++s)
            a0 = fmaf(sh_e[s], er[(size_t)s * ENC2], a0);
        ctx[(size_t)b * ENC2 + d] = a0 * inv;
    }
}

// ---------------------------------------------------------------------------
// Kernel 4 (per step): GRU cell via 6 WMMA tile-GEMMs.
// Block = 6 waves, one 16(batch) x 16(j) tile per wave:
//   waves 0..2 : ir/iz/in  = x @ W_ih[g*H+j0 .. ,:].T   (K = 1536)
//   waves 3..5 : hr/hz/hn  = h @ W_hh[g*H+j0 .. ,:].T   (K = 1024)
// x = [embedding[tok[b,t]], ctx[b]] gathered per lane; W rows are contiguous
// in K, so both A and B use 64-bit paired loads.  Combine gates in LDS.
// ---------------------------------------------------------------------------
__global__ void __launch_bounds__(192)
gru_kernel(const int* __restrict__ tokens, int t,
           const float* __restrict__ emb,  const float* __restrict__ ctx,
           const float* __restrict__ hprev,
           const float* __restrict__ Wih,  const float* __restrict__ Whh,
           const float* __restrict__ bih,  const float* __restrict__ bhh,
           float* __restrict__ hnew) {
    __shared__ float sh_t[6][256];

    const int lane = threadIdx.x & 31;
    const int w    = threadIdx.x >> 5;          // 0..5
    const int j0   = blockIdx.x * 16;
    const int nl   = lane & 15;                 // tile column (j)
    const int mb   = lane & 15;                 // tile row (batch) on A side
    const int koff = (lane < 16) ? 0 : 2;
    const int mh   = (lane < 16) ? 0 : 8;

    v8f acc = {};
    float bias;
    if (w < 3) {
        const int g = w;                                           // ir/iz/in
        const float* Wrow = Wih + (size_t)(g * H + j0 + nl) * XDIM;
        const float* embRow = emb + (size_t)tokens[mb * S + t] * EEMB;
        const float* ctxRow = ctx + (size_t)mb * ENC2;
        for (int k = 0; k < EEMB; k += 4) {
            f2 a = *(const f2*)(embRow + k + koff);
            f2 b = *(const f2*)(Wrow   + k + koff);
            acc = wmma4(a, b, acc);
        }
        for (int k = EEMB; k < XDIM; k += 4) {
            f2 a = *(const f2*)(ctxRow + (k - EEMB) + koff);
            f2 b = *(const f2*)(Wrow   + k + koff);
            acc = wmma4(a, b, acc);
        }
        bias = bih[g * H + j0 + nl];
    } else {
        const int g = w - 3;                                       // hr/hz/hn
        const float* Wrow = Whh + (size_t)(g * H + j0 + nl) * H;
        const float* hRow = hprev + (size_t)mb * H;
        for (int k = 0; k < H; k += 4) {
            f2 a = *(const f2*)(hRow + k + koff);
            f2 b = *(const f2*)(Wrow + k + koff);
            acc = wmma4(a, b, acc);
        }
        bias = bhh[g * H + j0 + nl];
    }
    for (int i = 0; i < 8; ++i)
        sh_t[w][(i + mh) * 16 + nl] = acc[i] + bias;
    __syncthreads();

    // gate combine: h' = (1-z)*n + z*h
    for (int idx = threadIdx.x; idx < 256; idx += 192) {
        const int b = idx >> 4, j = idx & 15;
        float ir = sh_t[0][idx], iz = sh_t[1][idx], in = sh_t[2][idx];
        float hr = sh_t[3][idx], hz = sh_t[4][idx], hn = sh_t[5][idx];
        float r = 1.0f / (1.0f + expf(-(ir + hr)));
        float z = 1.0f / (1.0f + expf(-(iz + hz)));
        float n = tanhf(in + r * hn);
        float ho = hprev[(size_t)b * H + j0 + j];
        hnew[(size_t)b * H + j0 + j] = (1.0f - z) * n + z * ho;
    }
}

// ---------------------------------------------------------------------------
// Kernel 5: logits = hseq @ W_out + b_out -> out[:,1:,:]
// M = 127*16 (one t per M-tile), K = 1024, N = 32000.  Dominant GEMM
// (133 GFLOP), f32 on V_WMMA_F32_16X16X4_F32.  Register-blocked 4x over M:
// each strided W_out load feeds 4 independent WMMAs, cutting B-operand L2
// traffic from ~16 GB to ~4 GB and keeping the matrix pipe busy.
// M-groups: ceil(127/4) = 32; the last group's 4th tile is computed from
// workspace scratch rows (safe) and its store is skipped (warp-uniform).
// ---------------------------------------------------------------------------
__global__ void __launch_bounds__(256)
logits_kernel(const float* __restrict__ hseq, const float* __restrict__ Wout,
              const float* __restrict__ bout, float* __restrict__ out) {
    const int lane = threadIdx.x & 31;
    const int warp = threadIdx.x >> 5;
    const int grp  = blockIdx.x * 8 + warp;       // 32 * 2000 groups
    const int mg   = grp / 2000;                  // 4 time steps per group
    const int tN   = grp % 2000;
    const int nl   = lane & 15;
    const int koff = (lane < 16) ? 0 : 2;
    const int n    = tN * 16 + nl;

    const float* A0 = hseq + (size_t)((mg * 4 + 0) * B + nl) * H;
    const float* A1 = hseq + (size_t)((mg * 4 + 1) * B + nl) * H;
    const float* A2 = hseq + (size_t)((mg * 4 + 2) * B + nl) * H;
    const float* A3 = hseq + (size_t)((mg * 4 + 3) * B + nl) * H;
    const float* Bcol = Wout + n;

    v8f c0 = {}, c1 = {}, c2 = {}, c3 = {};
    for (int k = 0; k < H; k += 4) {
        if (k + 8 < H) {  // warm L0 for the strided W_out rows 2 K-steps ahead
            __builtin_prefetch(Bcol + (size_t)(k + 8 + koff) * V, 0, 0);
        }
        f2 b = { Bcol[(size_t)(k + koff)     * V],
                 Bcol[(size_t)(k + koff + 1) * V] };
        c0 = wmma4(*(const f2*)(A0 + k + koff), b, c0);
        c1 = wmma4(*(const f2*)(A1 + k + koff), b, c1);
        c2 = wmma4(*(const f2*)(A2 + k + koff), b, c2);
        c3 = wmma4(*(const f2*)(A3 + k + koff), b, c3);
    }
    const float bias = bout[n];
    const int mh = (lane < 16) ? 0 : 8;
    for (int i = 0; i < 8; ++i) {
        const int b = i + mh;                     // batch row
        const size_t base = (size_t)b * S * V + n;
        out[base + (size_t)(mg * 4 + 1) * V] = c0[i] + bias;
        out[base + (size_t)(mg * 4 + 2) * V] = c1[i] + bias;
        out[base + (size_t)(mg * 4 + 3) * V] = c2[i] + bias;
        if (mg * 4 + 3 < TSTEPS)                  // warp-uniform guard
            out[base + (size_t)(mg * 4 + 4) * V] = c3[i] + bias;
    }
}

// ---------------------------------------------------------------------------
// Launcher
// ---------------------------------------------------------------------------
extern "C" void kernel_launch(void* const* d_in, const int* in_sizes, int n_in,
                              void* d_out, int out_size, void* d_ws, size_t ws_size,
                              hipStream_t stream) {
    (void)in_sizes; (void)n_in; (void)out_size; (void)ws_size;
    const int*   tokens  = (const int*)  d_in[0];
    const int*   lengths = (const int*)  d_in[1];
    const float* hidden0 = (const float*)d_in[2];
    const float* enc     = (const float*)d_in[3];
    const float* emb     = (const float*)d_in[4];
    const float* Wenc    = (const float*)d_in[5];
    const float* Wdec    = (const float*)d_in[6];
    const float* vattn   = (const float*)d_in[7];
    const float* Wih     = (const float*)d_in[8];
    const float* Whh     = (const float*)d_in[9];
    const float* bih     = (const float*)d_in[10];
    const float* bhh     = (const float*)d_in[11];
    const float* Wout    = (const float*)d_in[12];
    const float* bout    = (const float*)d_in[13];
    float* out = (float*)d_out;

    // workspace layout (floats): enc_proj | hseq | ctx (+ scratch after)
    float* encp = (float*)d_ws;                                   // 2048*1024
    float* hseq = encp + (size_t)B * S * A;                       // 127*16*1024
    float* ctx  = hseq + (size_t)TSTEPS * B * H;                  // 16*1024

    // enc_proj GEMM: 2048 warp-groups (4 M-tiles each), 8 waves/block
    encproj_kernel<<<256, 256, 0, stream>>>(enc, Wenc, encp);
    // out[:,0,:] one-hot
    row0_kernel<<<(B * V + 255) / 256, 256, 0, stream>>>(out);

    // sequential scan (stream order carries the recurrence)
    for (int t = 0; t < TSTEPS; ++t) {
        const float* hprev = (t == 0) ? hidden0 : (hseq + (size_t)(t - 1) * B * H);
        float*       hnew  = hseq + (size_t)t * B * H;
        attn_kernel<<<B, 256, 0, stream>>>(hprev, Wdec, vattn, encp, enc,
                                           lengths, ctx);
        gru_kernel<<<H / 16, 192, 0, stream>>>(tokens, t, emb, ctx, hprev,
                                               Wih, Whh, bih, bhh, hnew);
    }

    // logits GEMM: 32*2000 warp-groups, 8 waves/block -> 8000 blocks
    logits_kernel<<<8000, 256, 0, stream>>>(hseq, Wout, bout, out);
}